// GradedRelevanceLoss_7060926235084
// MI455X (gfx1250) — compile-verified
//
#include <hip/hip_runtime.h>

typedef float v2f __attribute__((ext_vector_type(2)));
typedef float v8f __attribute__((ext_vector_type(8)));

#define MARGIN      0.1f
#define RANK_WEIGHT 0.1f
#define CSPLIT      8     // column-range split for occupancy

// ---------------------------------------------------------------------------
// ws layout (d_ws): [0] float  : hinge sum over all ordered valid pairs
//                   [1] uint   : count of ordered valid pairs
// ---------------------------------------------------------------------------

static __device__ __forceinline__ unsigned grl_ballot32(bool pred) {
#if __has_builtin(__builtin_amdgcn_ballot_w32)
    return __builtin_amdgcn_ballot_w32(pred);
#else
    return (unsigned)__ballot(pred);
#endif
}

static __device__ __forceinline__ v8f grl_wmma_diff(v2f a, v2f b) {
    v8f z = {};
    // D = A x B : rank-2 product producing exact f32 pairwise differences
    return __builtin_amdgcn_wmma_f32_16x16x4_f32(false, a, false, b,
                                                 (short)0, z, false, false);
}

// Interior 16x16 tile: no bounds masking, only the exact t_diff != 0 test.
static __device__ __forceinline__ void grl_tile_interior(
        const float* __restrict__ p, const float* __restrict__ t,
        int j0, int sub, float selPos, float selNeg,
        v2f ap, v2f at, float& s, unsigned& c) {
    const int idx = j0 + sub;
    const float pj = p[idx];                 // unconditional, coalesced
    const float tj = t[idx];
    v2f bp, bt;
    bp.x = selPos;  bp.y = pj * selNeg;      // lanes 16-31 -> 0 (K2/K3 rows)
    bt.x = selPos;  bt.y = tj * selNeg;
    const v8f pd = grl_wmma_diff(ap, bp);    // p_m - p_j
    const v8f td = grl_wmma_diff(at, bt);    // t_m - t_j
#pragma unroll
    for (int r = 0; r < 8; ++r) {
        const float tdv = td[r];
        const float pdv = pd[r];
        // sp = sign(td)*pd via sign-bit transfer (td==0 corner masked below)
        const unsigned sgn = __float_as_uint(tdv) & 0x80000000u;
        const float sp = __uint_as_float(__float_as_uint(pdv) ^ sgn);
        const float h  = fmaxf(0.0f, MARGIN - sp);
        const bool  nz = (tdv != 0.0f);
        s += nz ? h : 0.0f;
        c += (unsigned)__popc(grl_ballot32(nz));   // SALU-side count
    }
}

// Edge 16x16 tile: full row/col masking (only touched when n % 16 != 0 etc).
static __device__ __forceinline__ void grl_tile_edge(
        const float* __restrict__ p, const float* __restrict__ t,
        int j0, int n, int sub, int rbase, float selPos, float selNeg,
        v2f ap, v2f at, float& s, unsigned& c) {
    const int col    = j0 + sub;
    const bool colOk = col < n;
    const int cc     = colOk ? col : (n - 1);
    const float pj = p[cc];
    const float tj = t[cc];
    v2f bp, bt;
    bp.x = selPos;  bp.y = pj * selNeg;
    bt.x = selPos;  bt.y = tj * selNeg;
    const v8f pd = grl_wmma_diff(ap, bp);
    const v8f td = grl_wmma_diff(at, bt);
#pragma unroll
    for (int r = 0; r < 8; ++r) {
        const float tdv = td[r];
        const float pdv = pd[r];
        const unsigned sgn = __float_as_uint(tdv) & 0x80000000u;
        const float sp = __uint_as_float(__float_as_uint(pdv) ^ sgn);
        const float h  = fmaxf(0.0f, MARGIN - sp);
        const bool valid = (tdv != 0.0f) && colOk && ((rbase + r) < n);
        s += valid ? h : 0.0f;
        c += (unsigned)__popc(grl_ballot32(valid));
    }
}

__global__ void grl_init_ws(float* ws) {
    if (threadIdx.x == 0) {
        ws[0] = 0.0f;
        ((unsigned int*)ws)[1] = 0u;
    }
}

// Pairwise ranking term. One wave = 16 rows; blockIdx.y owns a contiguous
// chunk of 16-wide column tiles.
__global__ __launch_bounds__(256) void grl_pair_kernel(
        const float* __restrict__ p, const float* __restrict__ t,
        float* __restrict__ ws, int n) {
    const int lane = threadIdx.x & 31;
    const int wid  = threadIdx.x >> 5;
    const int m0   = (blockIdx.x * 8 + wid) * 16;        // tile row base
    const int m0u  = __builtin_amdgcn_readfirstlane(m0); // force scalar/uniform
    const bool lowHalf = lane < 16;
    const int sub  = lane & 15;

    // Column-tile range owned by this blockIdx.y (all uniform arithmetic)
    const int tilesTotal = (n + 15) >> 4;
    const int tilesPer   = (tilesTotal + (int)gridDim.y - 1) / (int)gridDim.y;
    const int c0         = (int)blockIdx.y * tilesPer * 16;
    const int c1         = min(n, c0 + tilesPer * 16);
    int span = c1 - c0;  if (span < 0) span = 0;
    const int fullSpan = span & ~15;
    const int jFull    = c0 + fullSpan;

    // ---- A operands (16x4 f32): lanes 0-15 hold K0=value, K1=1; lanes 16-31 zero
    const int mrow = m0 + sub;
    const int mc   = mrow < n ? mrow : (n - 1);
    const float pm = p[mc];                 // unconditional clamped load
    const float tm = t[mc];
    const float selPos = lowHalf ?  1.0f : 0.0f;
    const float selNeg = lowHalf ? -1.0f : 0.0f;
    v2f ap, at;
    ap.x = pm * selPos;  ap.y = selPos;
    at.x = tm * selPos;  at.y = selPos;

    // Rows held by this lane's half of the C/D tile: rbase .. rbase+7
    const int rbase = m0 + (lowHalf ? 0 : 8);

    float    s = 0.0f;
    unsigned c = 0u;                        // meaningful on lane 0 only

    if ((m0u + 16) <= n) {                  // scalar condition -> s_cbranch
        int j0 = c0;
        for (; j0 + 32 <= jFull; j0 += 32) {         // x2 unrolled interior
            grl_tile_interior(p, t, j0,      sub, selPos, selNeg, ap, at, s, c);
            grl_tile_interior(p, t, j0 + 16, sub, selPos, selNeg, ap, at, s, c);
        }
        for (; j0 < jFull; j0 += 16)
            grl_tile_interior(p, t, j0, sub, selPos, selNeg, ap, at, s, c);
        if (fullSpan < span)                          // partial last column tile
            grl_tile_edge(p, t, jFull, n, sub, rbase, selPos, selNeg, ap, at, s, c);
    } else {
        for (int j0 = c0; j0 < c1; j0 += 16)          // partial row strip
            grl_tile_edge(p, t, j0, n, sub, rbase, selPos, selNeg, ap, at, s, c);
    }

    // ---- block reduction in LDS, one atomic pair per block
    __shared__ float    ssum[256];
    __shared__ unsigned scnt[256];
    ssum[threadIdx.x] = s;
    scnt[threadIdx.x] = (lane == 0) ? c : 0u;   // wave count counted once
    __syncthreads();
    for (int off = 128; off > 0; off >>= 1) {
        if ((int)threadIdx.x < off) {
            ssum[threadIdx.x] += ssum[threadIdx.x + off];
            scnt[threadIdx.x] += scnt[threadIdx.x + off];
        }
        __syncthreads();
    }
    if (threadIdx.x == 0) {
        atomicAdd(&ws[0], ssum[0]);
        atomicAdd(&((unsigned int*)ws)[1], scnt[0]);
    }
}

// O(n) MSE reduction + final combine (single block).
__global__ __launch_bounds__(256) void grl_finish_kernel(
        const float* __restrict__ p, const float* __restrict__ t,
        const float* __restrict__ ws, float* __restrict__ out, int n) {
    __shared__ float sm[256];
    float acc = 0.0f;
    for (int i = threadIdx.x; i < n; i += 256) {
        const float d = p[i] - t[i];
        acc += d * d;
    }
    sm[threadIdx.x] = acc;
    __syncthreads();
    for (int off = 128; off > 0; off >>= 1) {
        if ((int)threadIdx.x < off) sm[threadIdx.x] += sm[threadIdx.x + off];
        __syncthreads();
    }
    if (threadIdx.x == 0) {
        const float mse     = sm[0] / (float)n;
        const unsigned cnt  = ((const unsigned int*)ws)[1];
        const float ranking = (cnt > 0u) ? (ws[0] / (float)cnt) : 0.0f;
        out[0] = mse + RANK_WEIGHT * ranking;
    }
}

extern "C" void kernel_launch(void* const* d_in, const int* in_sizes, int n_in,
                              void* d_out, int out_size, void* d_ws, size_t ws_size,
                              hipStream_t stream) {
    const float* p = (const float*)d_in[0];
    const float* t = (const float*)d_in[1];
    float* out = (float*)d_out;
    float* ws  = (float*)d_ws;
    const int n = in_sizes[0];

    hipLaunchKernelGGL(grl_init_ws, dim3(1), dim3(64), 0, stream, ws);

    const int rowBlocks = (n + 127) / 128;   // 8 waves/block x 16 rows/wave
    hipLaunchKernelGGL(grl_pair_kernel, dim3(rowBlocks, CSPLIT), dim3(256), 0,
                       stream, p, t, ws, n);

    hipLaunchKernelGGL(grl_finish_kernel, dim3(1), dim3(256), 0, stream,
                       p, t, ws, out, n);
}